// QuantumFilter_65481071400027
// MI455X (gfx1250) — compile-verified
//
#include <hip/hip_runtime.h>
#include <hip/hip_bf16.h>
#include <math.h>

typedef float v2f __attribute__((ext_vector_type(2)));
typedef float v8f __attribute__((ext_vector_type(8)));

#define PI_F 3.14159265358979323846f

// ---------------------------------------------------------------------------
// Kernel 1: build B (16 x 64) from params.
// Simulate the parameterized part of the circuit on the 16 basis states to get
// U (16x16 complex), then A_i[k][c] = sum_b sgn_i(b) * (Ur[b][k]Ur[b][c] +
// Ui[b][k]Ui[b][c]); store B[k*64 + i*16 + c] = A_i[k][c].
// One wave (32 threads), threads 0..15 each evolve one basis column in LDS.
// ---------------------------------------------------------------------------
__global__ __launch_bounds__(32) void qf_build_B(const float* __restrict__ params,
                                                 float* __restrict__ Bws) {
    __shared__ float Ur[16][16];  // [basis b][column j]
    __shared__ float Ui[16][16];
    const int tid = threadIdx.x;

    if (tid < 16) {
        const int j = tid;
        for (int b = 0; b < 16; ++b) { Ur[b][j] = (b == j) ? 1.0f : 0.0f; Ui[b][j] = 0.0f; }

        // wire w occupies bit (3-w) of the flattened basis index
        auto apply_ry = [&](int w, float theta) {
            const int mask = 8 >> w;
            float c = cosf(0.5f * theta), s = sinf(0.5f * theta);
            for (int b = 0; b < 16; ++b) {
                if (b & mask) continue;
                const int b1 = b | mask;
                float r0 = Ur[b][j],  i0 = Ui[b][j];
                float r1 = Ur[b1][j], i1 = Ui[b1][j];
                Ur[b][j]  = c * r0 - s * r1;  Ui[b][j]  = c * i0 - s * i1;
                Ur[b1][j] = s * r0 + c * r1;  Ui[b1][j] = s * i0 + c * i1;
            }
        };
        auto apply_cnot = [&](int cw, int tw) {
            const int cmask = 8 >> cw, tmask = 8 >> tw;
            for (int b = 0; b < 16; ++b) {
                if (!(b & cmask) || (b & tmask)) continue;
                const int b1 = b | tmask;
                float r = Ur[b][j], im = Ui[b][j];
                Ur[b][j] = Ur[b1][j]; Ui[b][j] = Ui[b1][j];
                Ur[b1][j] = r;        Ui[b1][j] = im;
            }
        };
        auto apply_rz = [&](int w, float theta) {
            const int mask = 8 >> w;
            float c = cosf(0.5f * theta), s = sinf(0.5f * theta);
            for (int b = 0; b < 16; ++b) {
                float r = Ur[b][j], im = Ui[b][j];
                if (b & mask) { // multiply by conj(p) = c + i s
                    Ur[b][j] = r * c - im * s;  Ui[b][j] = im * c + r * s;
                } else {        // multiply by p = c - i s
                    Ur[b][j] = r * c + im * s;  Ui[b][j] = im * c - r * s;
                }
            }
        };

        for (int l = 0; l < 2; ++l) {
            const float* pl = params + l * 12;
            for (int w = 0; w < 4; ++w) apply_ry(w, pl[w * 3 + 0]);
            for (int cw = 0; cw < 3; ++cw) apply_cnot(cw, cw + 1);
            for (int w = 0; w < 4; ++w) apply_ry(w, pl[w * 3 + 1]);
            for (int cw = 0; cw < 3; ++cw) apply_cnot(cw + 1, cw);
            for (int w = 0; w < 4; ++w) apply_rz(w, pl[w * 3 + 2]);
        }
    }
    __syncthreads();

    // 1024 entries of B, 32 per thread
    for (int t = 0; t < 32; ++t) {
        const int e  = tid + 32 * t;
        const int i  = e >> 8;         // observable 0..3
        const int k  = (e >> 4) & 15;  // row of A_i
        const int cc = e & 15;         // col of A_i
        float acc = 0.0f;
        for (int b = 0; b < 16; ++b) {
            const float sgn = ((b >> (3 - i)) & 1) ? -1.0f : 1.0f;
            acc += sgn * (Ur[b][k] * Ur[b][cc] + Ui[b][k] * Ui[b][cc]);
        }
        Bws[k * 64 + i * 16 + cc] = acc;
    }
}

// ---------------------------------------------------------------------------
// Kernel 2: per-patch-location mean and 1/(std+1e-8) over batch*4 = 2048 elems
// 196 blocks x 256 threads.
// ---------------------------------------------------------------------------
__global__ __launch_bounds__(256) void qf_stats(const float* __restrict__ x,
                                                float* __restrict__ meanp,
                                                float* __restrict__ invp) {
    __shared__ float ssum[256];
    __shared__ float ssq[256];
    const int tid = threadIdx.x;
    const int p = blockIdx.x;
    const int pr = p / 14, pc = p - pr * 14;

    float sum = 0.0f, sq = 0.0f;
    for (int idx = tid; idx < 2048; idx += 256) {
        const int b = idx >> 2, e = idx & 3;
        const float v = x[b * 784 + (2 * pr + (e >> 1)) * 28 + 2 * pc + (e & 1)];
        sum += v; sq += v * v;
    }
    ssum[tid] = sum; ssq[tid] = sq;
    __syncthreads();
    for (int st = 128; st > 0; st >>= 1) {
        if (tid < st) { ssum[tid] += ssum[tid + st]; ssq[tid] += ssq[tid + st]; }
        __syncthreads();
    }
    if (tid == 0) {
        const float n = 2048.0f;
        const float m = ssum[0] / n;
        float var = (ssq[0] - n * m * m) / 2047.0f;   // ddof=1
        var = fmaxf(var, 0.0f);
        meanp[p] = m;
        invp[p] = 1.0f / (sqrtf(var) + 1e-8f);
    }
}

// ---------------------------------------------------------------------------
// Kernel 3: main fused kernel. One wave handles a tile of 16 samples.
//   S[m,k]   = prod over wires of cos/sin(theta_w/2)     (Kronecker vector)
//   T = S x B via 16x v_wmma_f32_16x16x4_f32 (K=16 in 4 steps, N=64 in 4 tiles)
//   z_i[m]   = sum_c T[m, 16i+c] * S[m,c]   (LDS transpose epilogue)
// 784 blocks x 256 threads (8 waves) -> 6272 tiles = 100352 samples.
// ---------------------------------------------------------------------------
__global__ __launch_bounds__(256) void qf_main(const float* __restrict__ x,
                                               const float* __restrict__ Bws,
                                               const float* __restrict__ meanp,
                                               const float* __restrict__ invp,
                                               float* __restrict__ out) {
    __shared__ float Dlds[8][4][16][16];   // [wave][observable][sample][col] = 32KB

    const int lane = threadIdx.x & 31;
    const int wv   = threadIdx.x >> 5;
    const int tile = blockIdx.x * 8 + wv;
    const int mloc = lane & 15;            // sample within tile
    const int hi   = lane >> 4;            // lane half (WMMA K split)
    const int m    = tile * 16 + mloc;     // global sample in [0, 100352)

    const int bb = m / 196;
    const int p  = m - bb * 196;
    const int pr = p / 14;
    const int pc = p - pr * 14;

    // B operands first (hide global-load latency under transcendental math):
    // 4 observables x 4 K-steps, 2 VGPRs each.
    // VGPR0: K = 4*kk + 2*hi, VGPR1: K = 4*kk + 2*hi + 1, N = i*16 + mloc.
    v2f bop[4][4];
#pragma unroll
    for (int i = 0; i < 4; ++i) {
#pragma unroll
        for (int kk = 0; kk < 4; ++kk) {
            const int k0  = 4 * kk + 2 * hi;
            const int col = i * 16 + mloc;
            bop[i][kk][0] = Bws[k0 * 64 + col];
            bop[i][kk][1] = Bws[(k0 + 1) * 64 + col];
        }
    }

    // angles -> half-angle sin/cos for the 4 wires (patch order (0,0)(0,1)(1,0)(1,1))
    const float mu = meanp[p];
    const float iv = invp[p] * (0.5f * PI_F);  // theta/2 = normalized * pi / 2
    const float* xb = x + bb * 784 + (2 * pr) * 28 + 2 * pc;
    float cw[4], sw[4];
#pragma unroll
    for (int e = 0; e < 4; ++e) {
        const float v = xb[(e >> 1) * 28 + (e & 1)];
        const float t = (v - mu) * iv;
        sw[e] = sinf(t);
        cw[e] = cosf(t);
    }

    // A operands built directly from the Kronecker factors.
    // Element k = 4*kk + 2*hi + j has bits: b3=kk&2 (wire0), b2=kk&1 (wire1),
    // b1=hi (wire2), b0=j (wire3).  Only wire2's factor depends on the lane
    // half -> a single v_cndmask instead of a register-array gather.
    const float e2 = hi ? sw[2] : cw[2];
    v2f aop[4];
#pragma unroll
    for (int kk = 0; kk < 4; ++kk) {
        const float f0 = (kk & 2) ? sw[0] : cw[0];
        const float f1 = (kk & 1) ? sw[1] : cw[1];
        const float g  = (f0 * f1) * e2;
        aop[kk][0] = g * cw[3];
        aop[kk][1] = g * sw[3];
    }

    // Full Kronecker vector for the epilogue (constant indices only).
    float s[16];
#pragma unroll
    for (int k = 0; k < 16; ++k) {
        const float f0 = (k & 8) ? sw[0] : cw[0];
        const float f1 = (k & 4) ? sw[1] : cw[1];
        const float f2 = (k & 2) ? sw[2] : cw[2];
        const float f3 = (k & 1) ? sw[3] : cw[3];
        s[k] = (f0 * f1) * (f2 * f3);
    }

    // 16x V_WMMA_F32_16X16X4_F32: D(16x16) accumulating over K=16 per observable
    v8f d[4];
#pragma unroll
    for (int i = 0; i < 4; ++i) {
        v8f c = {};
#pragma unroll
        for (int kk = 0; kk < 4; ++kk) {
            c = __builtin_amdgcn_wmma_f32_16x16x4_f32(
                    false, aop[kk], false, bop[i][kk], (short)0, c, false, false);
        }
        d[i] = c;
    }

    // Epilogue: transpose D through LDS so each lane dots its own sample's row
    // against its in-register s[].  D layout: VGPR r -> M = r + 8*hi, N = mloc.
#pragma unroll
    for (int i = 0; i < 4; ++i) {
#pragma unroll
        for (int r = 0; r < 8; ++r) {
            Dlds[wv][i][r + 8 * hi][mloc] = d[i][r];
        }
    }
    __syncthreads();

    // lanes 0-15 produce observables {0,2}, lanes 16-31 produce {1,3}
#pragma unroll
    for (int t = 0; t < 2; ++t) {
        const int i = hi + 2 * t;
        float z = 0.0f;
#pragma unroll
        for (int c = 0; c < 16; ++c) z += Dlds[wv][i][mloc][c] * s[c];
        // out[b, p*4 + i] = out[4*m + i]
        out[4 * m + i] = z;
    }
}

// ---------------------------------------------------------------------------
// Launch: workspace = B (1024 f32) | mean (196 f32) | inv (196 f32)
// ---------------------------------------------------------------------------
extern "C" void kernel_launch(void* const* d_in, const int* in_sizes, int n_in,
                              void* d_out, int out_size, void* d_ws, size_t ws_size,
                              hipStream_t stream) {
    const float* x      = (const float*)d_in[0];  // (512, 28, 28)
    const float* params = (const float*)d_in[1];  // (2, 4, 3)
    float* out = (float*)d_out;                   // (512, 784)

    float* ws    = (float*)d_ws;
    float* Bws   = ws;              // 1024 floats
    float* meanp = ws + 1024;       // 196 floats
    float* invp  = ws + 1024 + 196; // 196 floats

    qf_build_B<<<1, 32, 0, stream>>>(params, Bws);
    qf_stats<<<196, 256, 0, stream>>>(x, meanp, invp);
    qf_main<<<784, 256, 0, stream>>>(x, Bws, meanp, invp, out);
}